// W2V2Quantizer_28956669509848
// MI455X (gfx1250) — compile-verified
//
#include <hip/hip_runtime.h>
#include <math.h>

typedef __attribute__((ext_vector_type(2))) float v2f;
typedef __attribute__((ext_vector_type(8))) float v8f;

#define NTOK    65536      // B*T
#define DIMK    512
#define GV      640        // G*V
#define VNUM    320
#define VARD    128
#define TPB     64         // tokens per block (4 row tiles of 16)
#define NBLK    (NTOK / TPB)
#define XPAD    516        // x-tile row stride (floats): kills LDS bank conflicts
#define LDS_FLOATS (TPB * GV)              // logits tile is the larger union member
#define LDS_BYTES  (LDS_FLOATS * 4)        // 160 KB dynamic LDS

// ---------------------------------------------------------------------------
// Kernel 0: zero the code histogram (ws is poisoned; re-init every call)
// ---------------------------------------------------------------------------
__global__ void w2v2_zero(int* __restrict__ hist) {
    int t = blockIdx.x * blockDim.x + threadIdx.x;
    if (t < GV) hist[t] = 0;
}

// ---------------------------------------------------------------------------
// Kernel 1: fused GEMM (WMMA f32 16x16x4, 4 row-tiles share each B fragment)
//           + gumbel argmax + codebook gather + deterministic softmax partials
// ---------------------------------------------------------------------------
__global__ __launch_bounds__(256) void w2v2_main(
    const float* __restrict__ x,       // [NTOK, 512]
    const float* __restrict__ gum,     // [NTOK, 2, 320]
    const float* __restrict__ wproj,   // [512, 640] row-major
    const float* __restrict__ bproj,   // [640]
    const float* __restrict__ cb,      // [640, 128]
    float* __restrict__ out,           // [NTOK, 256]
    int*   __restrict__ hist,          // [640] global histogram
    float* __restrict__ pprob)         // [NBLK, 640] partial softmax sums
{
    // Dynamic LDS union: x-tile (64 rows x 516-float padded stride = 132 KB)
    // during GEMM, then logits tile (64 x 640 = 160 KB).
    extern __shared__ float ldsbuf[];
    __shared__ float lds_m[2 * TPB];     // per (token,group) max
    __shared__ float lds_inv[2 * TPB];   // per (token,group) 1/denominator

    const int tid       = threadIdx.x;
    const int tokenBase = blockIdx.x * TPB;

    // ---- stage x tile with padded row stride (8192 float4 / 256 threads) ----
    {
        const float4* xg = (const float4*)(x + (size_t)tokenBase * DIMK);
#pragma unroll
        for (int i = 0; i < 32; ++i) {
            const int idx = tid + i * 256;       // 0..8191
            const int row = idx >> 7;            // /128 float4 per row
            const int c4  = idx & 127;
            *(float4*)(ldsbuf + row * XPAD + c4 * 4) = xg[idx];
        }
    }
    __syncthreads();

    // ---- WMMA GEMM: wave owns 5 col tiles x 4 row tiles ---------------------
    const int wave  = tid >> 5;
    const int lane  = tid & 31;
    const int khalf = lane >> 4;     // 0: K=0,1   1: K=2,3 (f32 A/B layout)
    const int l16   = lane & 15;
    const int colBase0 = wave * 80;

    v8f acc[4][5];
#pragma unroll
    for (int rt = 0; rt < 4; ++rt)
#pragma unroll
        for (int j = 0; j < 5; ++j) acc[rt][j] = (v8f)0.0f;

#pragma unroll 2
    for (int k0 = 0; k0 < DIMK; k0 += 4) {
        const int kk = k0 + khalf * 2;
        v2f a[4];
#pragma unroll
        for (int rt = 0; rt < 4; ++rt) {
            const float* ar = ldsbuf + (rt * 16 + l16) * XPAD + kk;
            a[rt].x = ar[0];
            a[rt].y = ar[1];
        }
        const float* wr = wproj + (size_t)kk * GV;
#pragma unroll
        for (int j = 0; j < 5; ++j) {
            const int col = colBase0 + j * 16 + l16;
            v2f b;
            b.x = wr[col];
            b.y = wr[GV + col];
#pragma unroll
            for (int rt = 0; rt < 4; ++rt) {
                acc[rt][j] = __builtin_amdgcn_wmma_f32_16x16x4_f32(
                    false, a[rt], false, b, (short)0, acc[rt][j], false, false);
            }
        }
    }

    __syncthreads();   // all waves done reading x-tile before logits overwrite

    // ---- accumulators (+bias) -> LDS logits tile ----------------------------
#pragma unroll
    for (int j = 0; j < 5; ++j) {
        const int col  = colBase0 + j * 16 + l16;
        const float bv = bproj[col];
#pragma unroll
        for (int rt = 0; rt < 4; ++rt) {
#pragma unroll
            for (int r = 0; r < 8; ++r) {
                const int row = rt * 16 + r + khalf * 8;   // lanes 16-31 -> M+8
                ldsbuf[row * GV + col] = acc[rt][j][r] + bv;
            }
        }
    }
    __syncthreads();

    // ---- per (token, group): 2 lanes, each owns a contiguous half of V ------
    const int p   = tid >> 1;    // 0..127  (token t, group g)
    const int sub = tid & 1;
    const int t   = p >> 1;
    const int g   = p & 1;
    const size_t n = (size_t)tokenBase + t;

    const float* lrow = ldsbuf + t * GV + g * VNUM + sub * 160;
    const float4* g4  = (const float4*)(gum + n * (size_t)GV + g * VNUM + sub * 160);
    const int vbase   = sub * 160;

    float bestv = -3.4e38f; int besti  = 0;   // raw argmax (code_ppl)
    float bestg = -3.4e38f; int bestig = 0;   // logits+gumbel argmax (output)
#pragma unroll 4
    for (int i = 0; i < 40; ++i) {
        const float4 u4 = g4[i];
        const float4 f4 = *(const float4*)(lrow + i * 4);
        const float uu[4] = {u4.x, u4.y, u4.z, u4.w};
        const float ff[4] = {f4.x, f4.y, f4.z, f4.w};
#pragma unroll
        for (int q = 0; q < 4; ++q) {
            const int v = vbase + i * 4 + q;
            if (ff[q] > bestv) { bestv = ff[q]; besti = v; }
            const float gn = -logf(-logf(uu[q] + 1e-10f) + 1e-10f);
            const float fg = ff[q] + gn;
            if (fg > bestg) { bestg = fg; bestig = v; }
        }
    }
    {   // combine the two halves (lower index wins ties = first occurrence)
        const float ov = __shfl_xor(bestv, 1, 32);
        const int   oi = __shfl_xor(besti, 1, 32);
        if (ov > bestv || (ov == bestv && oi < besti)) { bestv = ov; besti = oi; }
        const float og  = __shfl_xor(bestg, 1, 32);
        const int   oig = __shfl_xor(bestig, 1, 32);
        if (og > bestg || (og == bestg && oig < bestig)) { bestg = og; bestig = oig; }
    }

    // stable softmax denominator
    float s = 0.0f;
#pragma unroll 4
    for (int i = 0; i < 160; ++i) s += expf(lrow[i] - bestv);
    s += __shfl_xor(s, 1, 32);

    if (sub == 0) {
        lds_m[p]   = bestv;
        lds_inv[p] = 1.0f / s;
        atomicAdd(&hist[g * VNUM + besti], 1);   // integer: deterministic
    }

    // straight-through output: winning codebook row (128 floats, 2 lanes)
    {
        const float4* cbr = (const float4*)(cb + (size_t)(g * VNUM + bestig) * VARD);
        float4* outr = (float4*)(out + n * (size_t)(2 * VARD) + g * VARD);
#pragma unroll
        for (int q = 0; q < 16; ++q) outr[sub * 16 + q] = cbr[sub * 16 + q];
    }
    __syncthreads();

    // ---- deterministic per-block softmax partial sums -----------------------
    for (int bin = tid; bin < GV; bin += 256) {
        const int gg = (bin >= VNUM) ? 1 : 0;
        float sum = 0.0f;
        for (int tt = 0; tt < TPB; ++tt) {
            const int pp = tt * 2 + gg;
            sum += expf(ldsbuf[tt * GV + bin] - lds_m[pp]) * lds_inv[pp];
        }
        pprob[(size_t)blockIdx.x * GV + bin] = sum;
    }
}

// ---------------------------------------------------------------------------
// Kernel 2: reduce partials -> avg_probs -> prob_ppl; histogram -> code_ppl
// ---------------------------------------------------------------------------
__global__ __launch_bounds__(256) void w2v2_final(
    const int* __restrict__ hist, const float* __restrict__ pprob,
    float* __restrict__ scal, int nblocks)
{
    __shared__ float ent[GV];
    __shared__ float cent[GV];
    const int tid = threadIdx.x;

    for (int bin = tid; bin < GV; bin += 256) {
        float sum = 0.0f;
        for (int b = 0; b < nblocks; ++b) sum += pprob[(size_t)b * GV + bin];
        const float ap = sum / (float)NTOK;
        ent[bin] = ap * logf(ap + 1e-7f);
        const float hp = (float)hist[bin] / (float)NTOK;
        cent[bin] = hp * logf(hp + 1e-7f);
    }
    __syncthreads();

    if (tid == 0) {
        float s0 = 0, s1 = 0, c0 = 0, c1 = 0;
        for (int v = 0; v < VNUM; ++v) {
            s0 += ent[v];  s1 += ent[VNUM + v];
            c0 += cent[v]; c1 += cent[VNUM + v];
        }
        scal[0] = expf(-c0) + expf(-c1);   // code_ppl
        scal[1] = expf(-s0) + expf(-s1);   // prob_ppl
    }
}

// ---------------------------------------------------------------------------
extern "C" void kernel_launch(void* const* d_in, const int* in_sizes, int n_in,
                              void* d_out, int out_size, void* d_ws, size_t ws_size,
                              hipStream_t stream) {
    const float* x     = (const float*)d_in[0];
    const float* gum   = (const float*)d_in[1];
    const float* wproj = (const float*)d_in[2];
    const float* bproj = (const float*)d_in[3];
    const float* cb    = (const float*)d_in[4];
    float* out = (float*)d_out;

    int*   hist  = (int*)d_ws;                          // 640 ints
    float* pprob = (float*)((char*)d_ws + 4096);        // NBLK*640 floats (~2.6 MB)

    w2v2_zero<<<2, 512, 0, stream>>>(hist);
    w2v2_main<<<NBLK, 256, LDS_BYTES, stream>>>(x, gum, wproj, bproj, cb, out, hist, pprob);
    w2v2_final<<<1, 256, 0, stream>>>(hist, pprob, out + (size_t)NTOK * 256, NBLK);
}